// Loss_6648609374713
// MI455X (gfx1250) — compile-verified
//
#include <hip/hip_runtime.h>
#include <hip/hip_bf16.h>
#include <math.h>

#define B_N 16384
#define D_DIM 1024
#define P_N (B_N / 2)
#define MARGIN_F 0.2f
#define TRADE_OFF_F 0.3f
#define EPS_COS_F 1e-8f

typedef float v2f __attribute__((ext_vector_type(2)));
typedef float v8f __attribute__((ext_vector_type(8)));

// ---------------- workspace accumulators ----------------
// ws[0] = sum of score[i, event[i]]          (CE)
// ws[1] = sum of (h_i - log(S_i+1e-6))*ev_i  (Cox numerator)
// ws[2] = sum of ev_i                        (Cox denominator)
// ws[3] = sum of relu(...)^2                 (contrastive)
__global__ void init_ws(float* ws) {
    if (threadIdx.x < 4) ws[threadIdx.x] = 0.0f;
}

// ---------------- CE ----------------
__global__ void ce_kernel(const float* __restrict__ score,
                          const int* __restrict__ ev, float* ws) {
    __shared__ float red[256];
    int i = blockIdx.x * 256 + threadIdx.x;
    red[threadIdx.x] = score[2 * i + ev[i]];
    __syncthreads();
    for (int s = 128; s > 0; s >>= 1) {
        if (threadIdx.x < s) red[threadIdx.x] += red[threadIdx.x + s];
        __syncthreads();
    }
    if (threadIdx.x == 0) atomicAdd(&ws[0], red[0]);
}

// ---------------- Cox (O(B^2) risk-set sums, LDS tiled) ----------------
#define COX_TILE 4096
__global__ void cox_kernel(const float* __restrict__ hazard,
                           const float* __restrict__ tme,
                           const int* __restrict__ ev, float* ws) {
    __shared__ float s_t[COX_TILE];
    __shared__ float s_e[COX_TILE];
    __shared__ float red[256];
    int i = blockIdx.x * 256 + threadIdx.x;
    float ti = tme[i];
    float hi = hazard[i];
    float S = 0.0f;
    for (int jt = 0; jt < B_N; jt += COX_TILE) {
        __syncthreads();
        for (int j = threadIdx.x; j < COX_TILE; j += 256) {
            s_t[j] = tme[jt + j];
            s_e[j] = expf(hazard[jt + j]);
        }
        __syncthreads();
#pragma unroll 8
        for (int j = 0; j < COX_TILE; ++j)
            S += (s_t[j] >= ti) ? s_e[j] : 0.0f;
    }
    float evf = (float)ev[i];
    float contrib = (hi - logf(S + 1e-6f)) * evf;

    red[threadIdx.x] = contrib;
    __syncthreads();
    for (int s = 128; s > 0; s >>= 1) {
        if (threadIdx.x < s) red[threadIdx.x] += red[threadIdx.x + s];
        __syncthreads();
    }
    if (threadIdx.x == 0) atomicAdd(&ws[1], red[0]);
    __syncthreads();
    red[threadIdx.x] = evf;
    __syncthreads();
    for (int s = 128; s > 0; s >>= 1) {
        if (threadIdx.x < s) red[threadIdx.x] += red[threadIdx.x + s];
        __syncthreads();
    }
    if (threadIdx.x == 0) atomicAdd(&ws[2], red[0]);
}

// ---------------- Contrastive: 6x6 Gram per pair via WMMA f32 16x16x4 ----
// One wave handles 2 pairs = 12 vectors packed into rows 0..11 of the A tile.
// B = A^T for a Gram product; the ISA layouts of A(16x4) and B(4x16) coincide
// (lane m holds K = 2*(lane>=16) .. +1 of row m&15), so SRC0 == SRC1.
// Rows 12..15 carry garbage (valid addresses) and are never read back.
// Two independent accumulator tiles break the WMMA->WMMA RAW chain so the
// scheduler can interleave two matrix-pipe dependency chains per wave.
__global__ void con_kernel(const float* __restrict__ Am,
                           const float* __restrict__ Bm,
                           const float* __restrict__ Cm,
                           const int* __restrict__ x1,
                           const int* __restrict__ x2, float* ws) {
    __shared__ float gram[8][256];
    const int lane = threadIdx.x & 31;
    const int wave = threadIdx.x >> 5;
    const int pairBase = blockIdx.x * 16 + wave * 2;
    const int r  = lane & 15;
    const int ko = (lane >> 4) << 1;   // K offset within 4-chunk: 0 or 2

    const float* rowPtr;
    if (r < 12) {
        int p   = pairBase + (r / 6);
        int rr  = r % 6;               // 0..5 -> a1,b1,c1,a2,b2,c2
        int idx = (rr < 3) ? x1[p] : x2[p];
        int vs  = rr % 3;
        const float* mat = (vs == 0) ? Am : (vs == 1) ? Bm : Cm;
        rowPtr = mat + (size_t)idx * D_DIM + ko;
    } else {
        rowPtr = Am + ko;              // harmless dummy row, result unread
    }

    v8f acc0 = {0.f, 0.f, 0.f, 0.f, 0.f, 0.f, 0.f, 0.f};
    v8f acc1 = {0.f, 0.f, 0.f, 0.f, 0.f, 0.f, 0.f, 0.f};
#pragma unroll 4
    for (int c = 0; c < D_DIM; c += 8) {
        v2f av0 = *(const v2f*)(rowPtr + c);
        v2f av1 = *(const v2f*)(rowPtr + c + 4);
        // D = A * A^T + C   (f32 in, f32 accumulate); two chains
        acc0 = __builtin_amdgcn_wmma_f32_16x16x4_f32(
            false, av0, false, av0, (short)0, acc0, false, false);
        acc1 = __builtin_amdgcn_wmma_f32_16x16x4_f32(
            false, av1, false, av1, (short)0, acc1, false, false);
    }
#pragma unroll
    for (int v = 0; v < 8; ++v) acc0[v] += acc1[v];

    // Spill the 16x16 f32 D tile: lane l, vgpr v holds D[v + 8*(l>=16)][l&15]
    const int mb = (lane >> 4) * 8;
    const int nn = lane & 15;
#pragma unroll
    for (int v = 0; v < 8; ++v) gram[wave][(mb + v) * 16 + nn] = acc0[v];
    __syncthreads();

    if (lane < 2) {
        const float* g = &gram[wave][0];
        const int o = lane * 6;        // pair 0 -> rows 0..5, pair 1 -> 6..11
        float inv[6];
#pragma unroll
        for (int k = 0; k < 6; ++k) {
            float nrm = sqrtf(g[(o + k) * 16 + (o + k)]);
            inv[k] = 1.0f / fmaxf(nrm, EPS_COS_F);
        }
#define G_(m_, n_) g[(o + (m_)) * 16 + (o + (n_))]
        float dxx = G_(0,1)*inv[0]*inv[1] + G_(0,2)*inv[0]*inv[2] + G_(1,2)*inv[1]*inv[2];
        float dyy = G_(3,4)*inv[3]*inv[4] + G_(3,5)*inv[3]*inv[5] + G_(4,5)*inv[4]*inv[5];
        float dxy = G_(0,3)*inv[0]*inv[3] + G_(1,4)*inv[1]*inv[4] + G_(2,5)*inv[2]*inv[5];
#undef G_
        float t = MARGIN_F + dxy - 0.5f * dxx - 0.5f * dyy;
        t = fmaxf(t, 0.0f);
        atomicAdd(&ws[3], t * t);
    }
}

// ---------------- final combine ----------------
__global__ void final_kernel(const float* __restrict__ ws, float* __restrict__ out) {
    float ce  = -ws[0] / (float)B_N;
    float cox = -ws[1] / (ws[2] + 1e-6f);
    float con =  ws[3] / (float)P_N;
    out[0] = ce + cox + TRADE_OFF_F * con;
}

extern "C" void kernel_launch(void* const* d_in, const int* in_sizes, int n_in,
                              void* d_out, int out_size, void* d_ws, size_t ws_size,
                              hipStream_t stream) {
    (void)in_sizes; (void)n_in; (void)out_size; (void)ws_size;
    const float* rep_a  = (const float*)d_in[0];
    const float* rep_b  = (const float*)d_in[1];
    const float* rep_c  = (const float*)d_in[2];
    const float* hazard = (const float*)d_in[3];
    const float* score  = (const float*)d_in[4];
    const float* tme    = (const float*)d_in[5];
    const int*   ev     = (const int*)d_in[6];
    const int*   x1     = (const int*)d_in[7];
    const int*   x2     = (const int*)d_in[8];
    float* ws  = (float*)d_ws;
    float* out = (float*)d_out;

    init_ws<<<1, 64, 0, stream>>>(ws);
    ce_kernel<<<B_N / 256, 256, 0, stream>>>(score, ev, ws);
    cox_kernel<<<B_N / 256, 256, 0, stream>>>(hazard, tme, ev, ws);
    con_kernel<<<P_N / 16, 256, 0, stream>>>(rep_a, rep_b, rep_c, x1, x2, ws);
    final_kernel<<<1, 1, 0, stream>>>(ws, out);
}